// RHIRL_69587060130271
// MI455X (gfx1250) — compile-verified
//
#include <hip/hip_runtime.h>
#include <hip/hip_bf16.h>
#include <math.h>

#define B_  2
#define N_  4096
#define H_  64
#define F2_ 128
#define EPS_ 1e-5f

typedef __attribute__((ext_vector_type(2))) float vf2;
typedef __attribute__((ext_vector_type(8))) float vf8;
typedef __attribute__((ext_vector_type(4))) unsigned vu4;
typedef __attribute__((ext_vector_type(8))) int     vi8;
typedef __attribute__((ext_vector_type(4))) int     vi4;

#if __has_builtin(__builtin_amdgcn_tensor_load_to_lds)
#define HAVE_TDM 1
#else
#define HAVE_TDM 0
#endif

// Full-precision CDNA5 WMMA: D(16x16 f32) = A(16x4 f32) * B(4x16 f32) + C
__device__ __forceinline__ vf8 wmma4(vf2 a, vf2 b, vf8 c) {
  return __builtin_amdgcn_wmma_f32_16x16x4_f32(false, a, false, b, (short)0, c,
                                               false, false);
}

#if HAVE_TDM
// 1D contiguous TDM copy: global -> LDS, nelem f32 elements (nelem < 65536).
// D# per CDNA5 ISA: group0 {count=1, lds_addr,
// global_addr, type=2}; group1 {data_size=4B, tensor_dim0=nelem,
// tensor_dim1=1, tile_dim0=nelem, tile_dim1=1, stride0=nelem}.
// 6-arg builtin form (this toolchain): (vu4, vi8, vi4, vi4, vi8, i32 cpol).
__device__ __forceinline__ void tdm_copy_1d(unsigned lds_byte_addr,
                                            const void* gsrc, unsigned nelem) {
  unsigned long long ga = (unsigned long long)(uintptr_t)gsrc;
  vu4 g0;
  g0[0] = 1u;                                             // count = 1
  g0[1] = lds_byte_addr;                                  // lds_addr
  g0[2] = (unsigned)(ga & 0xFFFFFFFFu);                   // global_addr[31:0]
  g0[3] = (unsigned)((ga >> 32) & 0x01FFFFFFu) | (2u << 30); // [56:32] | type=2
  vi8 g1;
  g1[0] = (int)(2u << 16);                                // data_size = 4B
  g1[1] = (int)((nelem & 0xFFFFu) << 16);                 // tensor_dim0 lo
  g1[2] = (int)(((nelem >> 16) & 0xFFFFu) | (1u << 16));  // dim0 hi | dim1=1
  g1[3] = (int)((nelem & 0xFFFFu) << 16);                 // tile_dim0 = nelem
  g1[4] = 1;                                              // tile_dim1=1, dim2=0
  g1[5] = (int)nelem;                                     // dim0_stride lo
  g1[6] = 0;
  g1[7] = 0;
  vi4 z4 = {0, 0, 0, 0};
  vi8 z8 = {0, 0, 0, 0, 0, 0, 0, 0};
  __builtin_amdgcn_tensor_load_to_lds(g0, g1, z4, z4, z8, 0);
}
#endif

// Stage a contiguous f32 tile into LDS: wave 0 drives the TDM (EXEC-agnostic,
// whole-wave branch), else cooperative vector loads.
__device__ __forceinline__ void stage_contig(float* dstLds, const float* src,
                                             int nelem, int tid, int nthreads) {
#if HAVE_TDM
  (void)nthreads;
  if (tid < 32) {
    tdm_copy_1d((unsigned)(uintptr_t)dstLds, src, (unsigned)nelem);
    __builtin_amdgcn_s_wait_tensorcnt(0);
  }
#else
  for (int q = tid * 4; q < nelem; q += nthreads * 4)
    *(float4*)(dstLds + q) = *(const float4*)(src + q);
#endif
}

// Async per-lane 16B copy global -> LDS (tracked by ASYNCcnt).
__device__ __forceinline__ void async_ld_b128(void* lds, const void* gsrc) {
  unsigned l = (unsigned)(uintptr_t)lds;
  unsigned long long g = (unsigned long long)(uintptr_t)gsrc;
  asm volatile("global_load_async_to_lds_b128 %0, %1, off"
               :: "v"(l), "v"(g) : "memory");
}
__device__ __forceinline__ void async_wait0() {
  asm volatile("s_wait_asynccnt 0x0" ::: "memory");
}

// ---------------------------------------------------------------------------
// C[64 x 64] per block = A[64 x K] * Bw[K x 64] + bias.
// grid.x = (B*N)/64, block = 128 (4 waves); wave w owns rows w*16..w*16+15 and
// all 64 cols (4 accumulators -> A fragment amortized 4x).
// ---------------------------------------------------------------------------
__global__ void k_gemm_n64(const float* __restrict__ A, int lda,
                           const float* __restrict__ Bw,
                           const float* __restrict__ bias,
                           float* __restrict__ C, int K) {
  __shared__ float As[64][68];
  __shared__ float Bs[64][64];     // unpadded: rows read lane-contiguous
  const int tid = threadIdx.x;
  const int wave = tid >> 5, lane = tid & 31;
  const long long r0 = (long long)blockIdx.x * 64;
  const int m = lane & 15;
  const int hi = lane >> 4;        // lanes 0-15: K pair {0,1}; 16-31: {2,3}
  const int mrow = wave * 16 + m;

  vf8 acc0 = {}, acc1 = {}, acc2 = {}, acc3 = {};

  for (int kc = 0; kc < K; kc += 64) {
    { // stage A tile 64x64 (each thread: 8 contiguous float4)
      int row = tid >> 1;
      int col = (tid & 1) * 32;
      const float* src = A + (r0 + row) * (long long)lda + kc + col;
#pragma unroll
      for (int q = 0; q < 8; ++q)
        *(float4*)&As[row][col + q * 4] = *(const float4*)(src + q * 4);
      if (kc + 64 < K) __builtin_prefetch(src + 64, 0, 0);
    }
    // stage B tile 64x64 via TDM (rows contiguous since ldb == 64)
    stage_contig(&Bs[0][0], Bw + (long long)kc * 64, 64 * 64, tid, 128);
    __syncthreads();
#pragma unroll
    for (int kk = 0; kk < 64; kk += 4) {
      vf2 a;
      a.x = As[mrow][kk + hi * 2 + 0];
      a.y = As[mrow][kk + hi * 2 + 1];
      vf2 b0, b1, b2, b3;
      b0.x = Bs[kk + hi * 2][m];      b0.y = Bs[kk + hi * 2 + 1][m];
      b1.x = Bs[kk + hi * 2][16 + m]; b1.y = Bs[kk + hi * 2 + 1][16 + m];
      b2.x = Bs[kk + hi * 2][32 + m]; b2.y = Bs[kk + hi * 2 + 1][32 + m];
      b3.x = Bs[kk + hi * 2][48 + m]; b3.y = Bs[kk + hi * 2 + 1][48 + m];
      acc0 = wmma4(a, b0, acc0);
      acc1 = wmma4(a, b1, acc1);
      acc2 = wmma4(a, b2, acc2);
      acc3 = wmma4(a, b3, acc3);
    }
    __syncthreads();
  }
#pragma unroll
  for (int v = 0; v < 8; ++v) {
    long long rr = r0 + wave * 16 + v + hi * 8;
    C[rr * 64 + m]      = acc0[v] + bias[m];
    C[rr * 64 + 16 + m] = acc1[v] + bias[16 + m];
    C[rr * 64 + 32 + m] = acc2[v] + bias[32 + m];
    C[rr * 64 + 48 + m] = acc3[v] + bias[48 + m];
  }
}

// ---------------------------------------------------------------------------
// BatchNorm over batch dim (B=2 => closed-form, elementwise) + ELU, in place.
// ---------------------------------------------------------------------------
__global__ void k_bn_elu_inplace(float* __restrict__ buf,
                                 const float* __restrict__ gamma,
                                 const float* __restrict__ beta, int M) {
  int f = blockIdx.x * blockDim.x + threadIdx.x;
  if (f >= M) return;
  float x0 = buf[f], x1 = buf[M + f];
  float mn = 0.5f * (x0 + x1);
  float d0 = x0 - mn, d1 = x1 - mn;
  float inv = rsqrtf(0.5f * (d0 * d0 + d1 * d1) + EPS_);
  float gg = gamma[f], bb = beta[f];
  float y0 = d0 * inv * gg + bb;
  float y1 = d1 * inv * gg + bb;
  buf[f]     = y0 > 0.f ? y0 : __expf(y0) - 1.f;
  buf[M + f] = y1 > 0.f ? y1 : __expf(y1) - 1.f;
}

// Same, but scatter the 64-wide features into h[:, :, 0:64] (row stride 128).
__global__ void k_bn_elu_to_h(const float* __restrict__ xs2,
                              const float* __restrict__ gamma,
                              const float* __restrict__ beta,
                              float* __restrict__ h, int M) {
  int f = blockIdx.x * blockDim.x + threadIdx.x;
  if (f >= M) return;
  float x0 = xs2[f], x1 = xs2[M + f];
  float mn = 0.5f * (x0 + x1);
  float d0 = x0 - mn, d1 = x1 - mn;
  float inv = rsqrtf(0.5f * (d0 * d0 + d1 * d1) + EPS_);
  float gg = gamma[f], bb = beta[f];
  float y0 = d0 * inv * gg + bb;
  float y1 = d1 * inv * gg + bb;
  int n = f >> 6, c = f & 63;
  long long o = (long long)n * F2_ + c;
  h[o]                       = y0 > 0.f ? y0 : __expf(y0) - 1.f;
  h[(long long)N_ * F2_ + o] = y1 > 0.f ? y1 : __expf(y1) - 1.f;
}

// ---------------------------------------------------------------------------
// Tiny "sp" path: y(2,4096) -> MLP(64) -> MLP(64), BN(B=2)+ELU each layer.
// ---------------------------------------------------------------------------
__global__ void k_sp_path(const float* __restrict__ y,
                          const float* __restrict__ W1, const float* __restrict__ b1,
                          const float* __restrict__ g1, const float* __restrict__ bt1,
                          const float* __restrict__ W2, const float* __restrict__ b2,
                          const float* __restrict__ g2, const float* __restrict__ bt2,
                          float* __restrict__ ys_out) {
  __shared__ float h1[2][64];
  int t = threadIdx.x;
  if (t < 64) {
    float a0 = b1[t], a1 = b1[t];
    for (int k = 0; k < N_; ++k) {
      float w = W1[k * 64 + t];
      a0 += y[k] * w;
      a1 += y[N_ + k] * w;
    }
    float mn = 0.5f * (a0 + a1), d0 = a0 - mn, d1 = a1 - mn;
    float inv = rsqrtf(0.5f * (d0 * d0 + d1 * d1) + EPS_);
    float v0 = d0 * inv * g1[t] + bt1[t];
    float v1 = d1 * inv * g1[t] + bt1[t];
    h1[0][t] = v0 > 0.f ? v0 : __expf(v0) - 1.f;
    h1[1][t] = v1 > 0.f ? v1 : __expf(v1) - 1.f;
  }
  __syncthreads();
  if (t < 64) {
    float a0 = b2[t], a1 = b2[t];
    for (int k = 0; k < 64; ++k) {
      float w = W2[k * 64 + t];
      a0 += h1[0][k] * w;
      a1 += h1[1][k] * w;
    }
    float mn = 0.5f * (a0 + a1), d0 = a0 - mn, d1 = a1 - mn;
    float inv = rsqrtf(0.5f * (d0 * d0 + d1 * d1) + EPS_);
    float v0 = d0 * inv * g2[t] + bt2[t];
    float v1 = d1 * inv * g2[t] + bt2[t];
    ys_out[t]      = v0 > 0.f ? v0 : __expf(v0) - 1.f;
    ys_out[64 + t] = v1 > 0.f ? v1 : __expf(v1) - 1.f;
  }
}

// Broadcast ys(2,64) into h[:, :, 64:128].
__global__ void k_bcast_ys(const float* __restrict__ ys, float* __restrict__ h) {
  int idx = blockIdx.x * blockDim.x + threadIdx.x;
  if (idx >= B_ * N_ * 64) return;
  int c = idx & 63;
  int n = (idx >> 6) & (N_ - 1);
  int b = idx >> 18;
  h[((long long)b * N_ + n) * F2_ + 64 + c] = ys[b * 64 + c];
}

// ---------------------------------------------------------------------------
// g = h @ Wg1 : M = B*N, K = 128, Nout = 128.  block = 256 (8 waves).
// ---------------------------------------------------------------------------
__global__ void k_gemm_n128(const float* __restrict__ A,
                            const float* __restrict__ Bw,
                            float* __restrict__ C) {
  __shared__ float As[16][132];
  __shared__ float Bs[64][128];    // unpadded: rows read lane-contiguous
  int tid = threadIdx.x, wave = tid >> 5, lane = tid & 31;
  long long r0 = (long long)blockIdx.x * 16;
  { // A tile 16x128, once
    int row = tid >> 4;
    int col = (tid & 15) * 8;
    const float* src = A + (r0 + row) * F2_ + col;
    *(float4*)&As[row][col]     = *(const float4*)(src);
    *(float4*)&As[row][col + 4] = *(const float4*)(src + 4);
  }
  int m = lane & 15, hi = lane >> 4, n0 = wave * 16;
  vf8 acc = {};
  for (int kc = 0; kc < 128; kc += 64) {
    // stage B chunk 64x128 via TDM (contiguous since ldb == 128)
    stage_contig(&Bs[0][0], Bw + (long long)kc * F2_, 64 * F2_, tid, 256);
    __syncthreads();
#pragma unroll
    for (int kk = 0; kk < 64; kk += 4) {
      vf2 a, b;
      a.x = As[m][kc + kk + hi * 2 + 0];
      a.y = As[m][kc + kk + hi * 2 + 1];
      b.x = Bs[kk + hi * 2 + 0][n0 + m];
      b.y = Bs[kk + hi * 2 + 1][n0 + m];
      acc = wmma4(a, b, acc);
    }
    __syncthreads();
  }
#pragma unroll
  for (int v = 0; v < 8; ++v)
    C[(r0 + v + hi * 8) * F2_ + n0 + m] = acc[v];
}

// si = g . a1, sj = g . a2 : one wave per row.
__global__ void k_sisj(const float* __restrict__ g, const float* __restrict__ a_vec,
                       float* __restrict__ si, float* __restrict__ sj) {
  int row = blockIdx.x * 8 + (threadIdx.x >> 5);
  int lane = threadIdx.x & 31;
  const float* gr = g + (long long)row * F2_;
  float s1 = 0.f, s2 = 0.f;
#pragma unroll
  for (int q = 0; q < 4; ++q) {
    int c = lane + q * 32;
    float gv = gr[c];
    s1 += gv * a_vec[c];
    s2 += gv * a_vec[F2_ + c];
  }
#pragma unroll
  for (int off = 16; off > 0; off >>= 1) {
    s1 += __shfl_down(s1, off, 32);
    s2 += __shfl_down(s2, off, 32);
  }
  if (lane == 0) { si[row] = s1; sj[row] = s2; }
}

// Masked-softmax row stats (max / sum / has-any). One block per row.
__global__ void k_softmax_stats(const float* __restrict__ si,
                                const float* __restrict__ sj,
                                const int* __restrict__ adj,
                                float* __restrict__ rmax, float* __restrict__ rsum,
                                float* __restrict__ rhas) {
  __shared__ float red[256];
  int row = blockIdx.x;                // b*N + i
  int b = row >> 12, i = row & (N_ - 1);
  int tid = threadIdx.x;
  float s_i = si[row];
  const float* sjb = sj + (long long)b * N_;
  const int* arow = adj + (long long)i * N_;

  float mx = -INFINITY;
  for (int j = tid; j < N_; j += 256) {
    if (arow[j] != 0) {
      float e = s_i + sjb[j];
      e = e > 0.f ? e : 0.2f * e;
      mx = fmaxf(mx, e);
    }
  }
  red[tid] = mx;
  __syncthreads();
  for (int s = 128; s > 0; s >>= 1) {
    if (tid < s) red[tid] = fmaxf(red[tid], red[tid + s]);
    __syncthreads();
  }
  mx = red[0];
  __syncthreads();

  float sum = 0.f;
  if (mx > -INFINITY) {
    for (int j = tid; j < N_; j += 256) {
      if (arow[j] != 0) {
        float e = s_i + sjb[j];
        e = e > 0.f ? e : 0.2f * e;
        sum += __expf(e - mx);
      }
    }
  }
  red[tid] = sum;
  __syncthreads();
  for (int s = 128; s > 0; s >>= 1) {
    if (tid < s) red[tid] += red[tid + s];
    __syncthreads();
  }
  if (tid == 0) {
    rmax[row] = mx;
    rsum[row] = red[0];
    rhas[row] = (mx > -INFINITY) ? 1.f : 0.f;
  }
}

// ---------------------------------------------------------------------------
// Flash-style attention + Wg2 + Wd + softsign, fused. One block per 16 rows,
// 128 threads (4 waves); wave w owns cols [w*32, w*32+32) -> 2 accumulators.
// g / Wg2 tiles staged with global_load_async_to_lds_b128 (ASYNCcnt path).
// ---------------------------------------------------------------------------
__global__ void k_attn_out(const float* __restrict__ gmat,
                           const float* __restrict__ si, const float* __restrict__ sj,
                           const float* __restrict__ rmax, const float* __restrict__ rsum,
                           const float* __restrict__ rhas, const int* __restrict__ adj,
                           const float* __restrict__ Wg2, const float* __restrict__ Wd,
                           const float* __restrict__ bd, float* __restrict__ out) {
  __shared__ float Ps[16][68];    // probability A-tile (16 x 64)
  __shared__ float Gs[64][128];   // g / Wg2 B-tile (unpadded)
  __shared__ float Hs[16][132];   // hp_att tile (A in phase 2 -> padded)
  __shared__ float Hs2[16][132];  // relu(hp @ Wg2) tile
  __shared__ float s_si[16], s_mx[16], s_rinv[16];

  int tid = threadIdx.x, wave = tid >> 5, lane = tid & 31;
  long long r0 = (long long)blockIdx.x * 16;   // global row = b*N + i
  int b = (int)(r0 >> 12);
  int i0 = (int)(r0 & (N_ - 1));

  if (tid < 16) {
    int r = (int)r0 + tid;
    s_si[tid] = si[r];
    s_mx[tid] = rmax[r];
    float rs = rsum[r];
    s_rinv[tid] = (rhas[r] > 0.5f && rs > 0.f) ? 1.f / rs : 0.f;
  }
  __syncthreads();

  int m = lane & 15, hi = lane >> 4, n0 = wave * 32;
  vf8 accA = {}, accB = {};
  const float* sjb = sj + (long long)b * N_;
  const float* gb = gmat + (long long)b * N_ * F2_;

  for (int j0 = 0; j0 < N_; j0 += 64) {
#pragma unroll
    for (int q = 0; q < 8; ++q) {          // probabilities tile 16x64
      int idx = tid + q * 128;
      int pm = idx >> 6, jj = idx & 63;
      int j = j0 + jj;
      float p = 0.f;
      if (adj[(long long)(i0 + pm) * N_ + j] != 0) {
        float e = s_si[pm] + sjb[j];
        e = e > 0.f ? e : 0.2f * e;
        p = __expf(e - s_mx[pm]) * s_rinv[pm];
      }
      Ps[pm][jj] = p;
    }
#pragma unroll
    for (int q = 0; q < 16; ++q) {         // g tile 64x128, async -> LDS
      int idx = tid + q * 128;
      int grow = idx >> 5, gcol = (idx & 31) * 4;
      async_ld_b128(&Gs[grow][gcol], gb + (long long)(j0 + grow) * F2_ + gcol);
    }
    async_wait0();
    __syncthreads();
#pragma unroll
    for (int kk = 0; kk < 64; kk += 4) {
      vf2 a, b0, b1;
      a.x = Ps[m][kk + hi * 2 + 0];
      a.y = Ps[m][kk + hi * 2 + 1];
      b0.x = Gs[kk + hi * 2 + 0][n0 + m];      b0.y = Gs[kk + hi * 2 + 1][n0 + m];
      b1.x = Gs[kk + hi * 2 + 0][n0 + 16 + m]; b1.y = Gs[kk + hi * 2 + 1][n0 + 16 + m];
      accA = wmma4(a, b0, accA);
      accB = wmma4(a, b1, accB);
    }
    __syncthreads();
  }
#pragma unroll
  for (int v = 0; v < 8; ++v) {
    Hs[v + hi * 8][n0 + m]      = accA[v];
    Hs[v + hi * 8][n0 + 16 + m] = accB[v];
  }
  __syncthreads();

  // hp2 = relu(Hs @ Wg2), K = 128 staged in two 64-row async chunks
  vf8 acc2A = {}, acc2B = {};
  for (int kc = 0; kc < 128; kc += 64) {
#pragma unroll
    for (int q = 0; q < 16; ++q) {
      int idx = tid + q * 128;
      int wrow = idx >> 5, wcol = (idx & 31) * 4;
      async_ld_b128(&Gs[wrow][wcol], Wg2 + (long long)(kc + wrow) * F2_ + wcol);
    }
    async_wait0();
    __syncthreads();
#pragma unroll
    for (int kk = 0; kk < 64; kk += 4) {
      vf2 a, b0, b1;
      a.x = Hs[m][kc + kk + hi * 2 + 0];
      a.y = Hs[m][kc + kk + hi * 2 + 1];
      b0.x = Gs[kk + hi * 2 + 0][n0 + m];      b0.y = Gs[kk + hi * 2 + 1][n0 + m];
      b1.x = Gs[kk + hi * 2 + 0][n0 + 16 + m]; b1.y = Gs[kk + hi * 2 + 1][n0 + 16 + m];
      acc2A = wmma4(a, b0, acc2A);
      acc2B = wmma4(a, b1, acc2B);
    }
    __syncthreads();
  }
#pragma unroll
  for (int v = 0; v < 8; ++v) {
    Hs2[v + hi * 8][n0 + m]      = fmaxf(acc2A[v], 0.f);
    Hs2[v + hi * 8][n0 + 16 + m] = fmaxf(acc2B[v], 0.f);
  }
  __syncthreads();

  if (tid < 16) {
    float o = bd[0];
    for (int c = 0; c < F2_; ++c) o += Hs2[tid][c] * Wd[c];
    out[r0 + tid] = o / (1.f + fabsf(o)) - 1.f;
  }
}

// ---------------------------------------------------------------------------
extern "C" void kernel_launch(void* const* d_in, const int* in_sizes, int n_in,
                              void* d_out, int out_size, void* d_ws, size_t ws_size,
                              hipStream_t stream) {
  (void)in_sizes; (void)n_in; (void)out_size; (void)ws_size;
  const float* x      = (const float*)d_in[0];
  const float* y      = (const float*)d_in[1];
  const int*   adj    = (const int*)d_in[2];
  const float* W1_seg = (const float*)d_in[3];
  const float* b1_seg = (const float*)d_in[4];
  const float* g1_seg = (const float*)d_in[5];
  const float* bt1_seg= (const float*)d_in[6];
  const float* W2_seg = (const float*)d_in[7];
  const float* b2_seg = (const float*)d_in[8];
  const float* g2_seg = (const float*)d_in[9];
  const float* bt2_seg= (const float*)d_in[10];
  const float* W1_sp  = (const float*)d_in[11];
  const float* b1_sp  = (const float*)d_in[12];
  const float* g1_sp  = (const float*)d_in[13];
  const float* bt1_sp = (const float*)d_in[14];
  const float* W2_sp  = (const float*)d_in[15];
  const float* b2_sp  = (const float*)d_in[16];
  const float* g2_sp  = (const float*)d_in[17];
  const float* bt2_sp = (const float*)d_in[18];
  const float* Wg1    = (const float*)d_in[19];
  const float* Wg2    = (const float*)d_in[20];
  const float* a_vec  = (const float*)d_in[21];
  const float* Wd     = (const float*)d_in[22];
  const float* bd     = (const float*)d_in[23];

  float* ws  = (float*)d_ws;
  float* xs  = ws;                  // B*N*64   = 524288
  float* xs2 = ws + 524288;         // B*N*64
  float* h   = ws + 1048576;        // B*N*128  = 1048576
  float* g   = ws + 2097152;        // B*N*128
  float* ys  = ws + 3145728;        // 128
  float* si  = ws + 3145856;        // 8192
  float* sj  = ws + 3154048;        // 8192
  float* rmx = ws + 3162240;        // 8192
  float* rsm = ws + 3170432;        // 8192
  float* rhs = ws + 3178624;        // 8192

  // xs = x @ W1_seg + b1 ; BN+ELU (in place)
  k_gemm_n64<<<(B_ * N_) / 64, 128, 0, stream>>>(x, N_, W1_seg, b1_seg, xs, N_);
  k_bn_elu_inplace<<<(N_ * H_ + 255) / 256, 256, 0, stream>>>(xs, g1_seg, bt1_seg, N_ * H_);

  // xs2 = xs @ W2_seg + b2 ; BN+ELU -> h[:, :, 0:64]
  k_gemm_n64<<<(B_ * N_) / 64, 128, 0, stream>>>(xs, 64, W2_seg, b2_seg, xs2, 64);
  k_bn_elu_to_h<<<(N_ * 64 + 255) / 256, 256, 0, stream>>>(xs2, g2_seg, bt2_seg, h, N_ * 64);

  // sp path -> ys ; broadcast into h[:, :, 64:128]
  k_sp_path<<<1, 64, 0, stream>>>(y, W1_sp, b1_sp, g1_sp, bt1_sp,
                                  W2_sp, b2_sp, g2_sp, bt2_sp, ys);
  k_bcast_ys<<<(B_ * N_ * 64 + 255) / 256, 256, 0, stream>>>(ys, h);

  // g = h @ Wg1 ; attention scores
  k_gemm_n128<<<(B_ * N_) / 16, 256, 0, stream>>>(h, Wg1, g);
  k_sisj<<<(B_ * N_) / 8, 256, 0, stream>>>(g, a_vec, si, sj);
  k_softmax_stats<<<B_ * N_, 256, 0, stream>>>(si, sj, adj, rmx, rsm, rhs);

  // fused: att@g -> relu(.@Wg2) -> .@Wd + bd -> softsign - 1
  k_attn_out<<<(B_ * N_) / 16, 128, 0, stream>>>(g, si, sj, rmx, rsm, rhs, adj,
                                                 Wg2, Wd, bd, (float*)d_out);
}